// MRP_75514114998797
// MI455X (gfx1250) — compile-verified
//
#include <hip/hip_runtime.h>
#include <math.h>

typedef float v2f __attribute__((ext_vector_type(2)));
typedef float v8f __attribute__((ext_vector_type(8)));
typedef int v4i_t __attribute__((__vector_size__(16)));

#define T_TOT 262144
#define K_ST  32
#define D_DIM 64
#define L_CH  64
#define P_CH  (T_TOT / L_CH)   /* 4096 chunks */
#define WPB   4                /* waves (chunks) per block */

#define AS1 __attribute__((address_space(1)))
#define AS3 __attribute__((address_space(3)))

__device__ __forceinline__ float redMax32(float v) {
#pragma unroll
  for (int m = 16; m >= 1; m >>= 1) v = fmaxf(v, __shfl_xor(v, m, 32));
  return v;
}
__device__ __forceinline__ float redSum32(float v) {
#pragma unroll
  for (int m = 16; m >= 1; m >>= 1) v += __shfl_xor(v, m, 32);
  return v;
}

__device__ __forceinline__ v8f wmma4(v2f a, v2f b, v8f c) {
  // D = A(16x4,f32) * B(4x16,f32) + C(16x16,f32)
  return __builtin_amdgcn_wmma_f32_16x16x4_f32(false, a, false, b, (short)0, c,
                                               false, false);
}

__device__ __forceinline__ void waitAsync0() {
#if __has_builtin(__builtin_amdgcn_s_wait_asynccnt)
  __builtin_amdgcn_s_wait_asynccnt(0);
#else
  asm volatile("s_wait_asynccnt 0" ::: "memory");
#endif
}

// ---------------------------------------------------------------------------
// Kernel 1: precompute softmaxes, Gamma params, fused emission weights W[128][32]
// ws layout (floats): W[4096] | expA[1024] | pi0[32] | ga[32] | gb[32] | cst[32]
// ---------------------------------------------------------------------------
__global__ __launch_bounds__(32) void precompute_kernel(
    const float* __restrict__ A, const float* __restrict__ log_ab,
    const float* __restrict__ mu, const float* __restrict__ log_sigma,
    const float* __restrict__ pi, float* __restrict__ wsW,
    float* __restrict__ wsExpA, float* __restrict__ wsPi0,
    float* __restrict__ wsGa, float* __restrict__ wsGb,
    float* __restrict__ wsCst) {
  const int lane = threadIdx.x & 31;

  // pi0 = log_softmax(pi)
  float pv = pi[lane];
  float pm = redMax32(pv);
  float ps = redSum32(__expf(pv - pm));
  wsPi0[lane] = pv - pm - logf(ps);

  // expA = softmax(A, axis=1)  (exp of log_softmax rows)
  for (int i = 0; i < 32; ++i) {
    float v = A[i * 32 + lane];
    float rm = redMax32(v);
    float e = __expf(v - rm);
    float rs = redSum32(e);
    wsExpA[i * 32 + lane] = e / rs;
  }

  // Gamma(conc=a, rate=b) constants
  float la = log_ab[lane * 2 + 0];
  float lb = log_ab[lane * 2 + 1];
  float a = expf(la), b = expf(lb);
  wsGa[lane] = a;
  wsGb[lane] = b;
  float cst = a * lb - lgammaf(a);  // a*log(b) - lgamma(a)

  // Gaussian: W[d][j] = -0.5*inv_var ; W[64+d][j] = mu*inv_var
  float sls = 0.f, smm = 0.f;
  for (int d = 0; d < D_DIM; ++d) {
    float ls = log_sigma[lane * D_DIM + d];
    float iv = expf(-2.f * ls);
    float md = mu[lane * D_DIM + d];
    wsW[d * 32 + lane] = -0.5f * iv;
    wsW[(64 + d) * 32 + lane] = md * iv;
    sls += ls;
    smm += md * md * iv;
  }
  // const = -sum(log_sigma) - 0.5*D*log(2pi) - 0.5*sum(mu^2*inv_var) + gamma const
  cst += -sls - 0.5f * smm - 32.0f * 1.8378770664093453f;
  wsCst[lane] = cst;
}

// ---------------------------------------------------------------------------
// Kernel 2: per-chunk scaled-linear transfer matrix via WMMA f32 16x16x4
// One wave per chunk of L_CH timesteps. Chunk 0 skips t==0 (handled in final).
// ---------------------------------------------------------------------------
__global__ __launch_bounds__(WPB * 32) void chunk_kernel(
    const float* __restrict__ Y, const float* __restrict__ dT,
    const float* __restrict__ wsW, const float* __restrict__ wsExpA,
    const float* __restrict__ wsGa, const float* __restrict__ wsGb,
    const float* __restrict__ wsCst, float* __restrict__ Mout,
    float* __restrict__ Sout) {
  __shared__ __align__(16) float sW[128 * 32];        // emission weights (block-shared)
  __shared__ __align__(16) float sY[WPB][16 * 64];    // 16 timesteps of Y
  __shared__ __align__(16) float sC[WPB][32 * 32];    // running product, row-major
  __shared__ __align__(16) float sLL[WPB][16 * 32];   // y_ll tile, [t][j]
  __shared__ float sDT[WPB][16];

  const int lane = threadIdx.x & 31;
  const int w = threadIdx.x >> 5;
  const int chunk = blockIdx.x * WPB + w;
  const int half = lane >> 4;   // lane group 0/1
  const int l15 = lane & 15;
  const int dlt = half * 2;     // K offset within group of 4

  // stage W (block-cooperative)
  for (int i = threadIdx.x; i < 128 * 32 / 4; i += blockDim.x)
    ((float4*)sW)[i] = ((const float4*)wsW)[i];
  __syncthreads();

  const float ga = wsGa[lane], gb = wsGb[lane];
  const float cst = wsCst[lane];
  const float cst_lo = __shfl(cst, l15, 32);
  const float cst_hi = __shfl(cst, 16 + l15, 32);

  // expA resident in registers, B-layout: b0 holds rows {4kk+dlt}, b1 rows {4kk+1+dlt}
  float eA0[8][2], eA1[8][2];
#pragma unroll
  for (int kk = 0; kk < 8; ++kk)
#pragma unroll
    for (int jb = 0; jb < 2; ++jb) {
      eA0[kk][jb] = wsExpA[(4 * kk + dlt) * 32 + jb * 16 + l15];
      eA1[kk][jb] = wsExpA[(4 * kk + 1 + dlt) * 32 + jb * 16 + l15];
    }

  // C = Identity
  for (int r = 0; r < 32; ++r) sC[w][r * 32 + lane] = (r == lane) ? 1.f : 0.f;

  float S = 0.f;     // accumulated log-scale
  float winv = 1.f;  // pending renorm folded into next step's w
  float lmx = 0.f;   // log of pending renorm

  const int tbase = chunk * L_CH;
  for (int g = 0; g < L_CH / 16; ++g) {
    const int t0 = tbase + g * 16;
    // stage Y[t0..t0+15][0..63]  (async DMA to LDS when available)
#if __has_builtin(__builtin_amdgcn_global_load_async_to_lds_b128)
#pragma unroll
    for (int i = 0; i < 8; ++i) {
      int flat = i * 128 + lane * 4;
      __builtin_amdgcn_global_load_async_to_lds_b128(
          (AS1 v4i_t*)&Y[(size_t)t0 * 64 + flat], (AS3 v4i_t*)&sY[w][flat], 0,
          0);
    }
    waitAsync0();
    asm volatile("" ::: "memory");
#else
#pragma unroll
    for (int i = 0; i < 8; ++i) {
      int flat = i * 128 + lane * 4;
      *(float4*)&sY[w][flat] = *(const float4*)&Y[(size_t)t0 * 64 + flat];
    }
#endif
    if (lane < 16) sDT[w][lane] = dT[t0 + lane];

    // y_ll tile (16 x 32) = [Y^2 | Y] @ W  + cst, via WMMA f32 16x16x4
    v8f acc0 = {cst_lo, cst_lo, cst_lo, cst_lo, cst_lo, cst_lo, cst_lo, cst_lo};
    v8f acc1 = {cst_hi, cst_hi, cst_hi, cst_hi, cst_hi, cst_hi, cst_hi, cst_hi};
#pragma unroll
    for (int kk = 0; kk < 32; ++kk) {
      int col = 4 * (kk & 15) + dlt;
      v2f a = *(v2f*)&sY[w][l15 * 64 + col];
      if (kk < 16) a = a * a;  // first 64 K-cols are Y*Y
      v2f b0 = {sW[(4 * kk + dlt) * 32 + l15], sW[(4 * kk + 1 + dlt) * 32 + l15]};
      v2f b1 = {sW[(4 * kk + dlt) * 32 + 16 + l15],
                sW[(4 * kk + 1 + dlt) * 32 + 16 + l15]};
      acc0 = wmma4(a, b0, acc0);
      acc1 = wmma4(a, b1, acc1);
    }
#pragma unroll
    for (int v = 0; v < 8; ++v) {
      int t = v + half * 8;
      sLL[w][t * 32 + l15] = acc0[v];
      sLL[w][t * 32 + 16 + l15] = acc1[v];
    }

    // sequential scan over the 16 timesteps (chunk 0 starts at t=1)
    const int tt0 = (t0 == 0) ? 1 : 0;
    for (int tt = tt0; tt < 16; ++tt) {
      float dt = sDT[w][tt];
      float ldt = __logf(dt);
      float ll = sLL[w][tt * 32 + lane] + ldt * (ga - 1.f) - dt * gb;
      float m = redMax32(ll);
      float wv = __expf(ll - m) * winv;
      S += m + lmx;
      float wlo = __shfl(wv, l15, 32);
      float whi = __shfl(wv, 16 + l15, 32);

      float mx = 0.f;
      // phase ib=0: reads C rows 0..15, writes rows 0..15 (no overlap w/ ib=1 reads)
#pragma unroll
      for (int ib = 0; ib < 2; ++ib) {
        v8f t_0 = {0.f, 0.f, 0.f, 0.f, 0.f, 0.f, 0.f, 0.f};
        v8f t_1 = {0.f, 0.f, 0.f, 0.f, 0.f, 0.f, 0.f, 0.f};
#pragma unroll
        for (int kk = 0; kk < 8; ++kk) {
          v2f a = *(v2f*)&sC[w][(ib * 16 + l15) * 32 + 4 * kk + dlt];
          v2f b0 = {eA0[kk][0], eA1[kk][0]};
          v2f b1 = {eA0[kk][1], eA1[kk][1]};
          t_0 = wmma4(a, b0, t_0);
          t_1 = wmma4(a, b1, t_1);
        }
#pragma unroll
        for (int v = 0; v < 8; ++v) {
          float x0 = t_0[v] * wlo;
          float x1 = t_1[v] * whi;
          mx = fmaxf(mx, fmaxf(x0, x1));
          int r = ib * 16 + v + half * 8;
          sC[w][r * 32 + l15] = x0;
          sC[w][r * 32 + 16 + l15] = x1;
        }
      }
      mx = redMax32(mx);
      mx = fmaxf(mx, 1e-37f);
      winv = 1.0f / mx;   // folded into next step's w
      lmx = __logf(mx);
    }
  }

  // write chunk matrix + scale (stored matrix * exp(S) = true product)
  for (int r = 0; r < 32; ++r)
    Mout[(size_t)chunk * 1024 + r * 32 + lane] = sC[w][r * 32 + lane];
  if (lane == 0) Sout[chunk] = S;
}

// ---------------------------------------------------------------------------
// Kernel 3: pairwise product of chunk matrices (time order), WMMA + renorm
// ---------------------------------------------------------------------------
__global__ __launch_bounds__(32) void reduce_kernel(
    const float* __restrict__ Min, const float* __restrict__ Sin,
    float* __restrict__ Mout, float* __restrict__ Sout) {
  __shared__ __align__(16) float sA[1024];
  const int lane = threadIdx.x & 31;
  const int half = lane >> 4;
  const int l15 = lane & 15;
  const int dlt = half * 2;
  const int p = blockIdx.x;

  const float* Am = Min + (size_t)(2 * p) * 1024;
  const float* Bm = Min + (size_t)(2 * p + 1) * 1024;
#pragma unroll
  for (int i = 0; i < 8; ++i)
    ((float4*)sA)[i * 32 + lane] = ((const float4*)Am)[i * 32 + lane];

  float b0r[8][2], b1r[8][2];
#pragma unroll
  for (int kk = 0; kk < 8; ++kk)
#pragma unroll
    for (int jb = 0; jb < 2; ++jb) {
      b0r[kk][jb] = Bm[(4 * kk + dlt) * 32 + jb * 16 + l15];
      b1r[kk][jb] = Bm[(4 * kk + 1 + dlt) * 32 + jb * 16 + l15];
    }

  v8f t[2][2];
  float mx = 0.f;
#pragma unroll
  for (int ib = 0; ib < 2; ++ib) {
    v8f a0 = {0.f, 0.f, 0.f, 0.f, 0.f, 0.f, 0.f, 0.f};
    v8f a1 = {0.f, 0.f, 0.f, 0.f, 0.f, 0.f, 0.f, 0.f};
#pragma unroll
    for (int kk = 0; kk < 8; ++kk) {
      v2f a = *(v2f*)&sA[(ib * 16 + l15) * 32 + 4 * kk + dlt];
      v2f b0 = {b0r[kk][0], b1r[kk][0]};
      v2f b1 = {b0r[kk][1], b1r[kk][1]};
      a0 = wmma4(a, b0, a0);
      a1 = wmma4(a, b1, a1);
    }
    t[ib][0] = a0;
    t[ib][1] = a1;
#pragma unroll
    for (int v = 0; v < 8; ++v) mx = fmaxf(mx, fmaxf(a0[v], a1[v]));
  }
  mx = redMax32(mx);
  mx = fmaxf(mx, 1e-37f);
  float inv = 1.0f / mx;
#pragma unroll
  for (int ib = 0; ib < 2; ++ib)
#pragma unroll
    for (int v = 0; v < 8; ++v) {
      int r = ib * 16 + v + half * 8;
      Mout[(size_t)p * 1024 + r * 32 + l15] = t[ib][0][v] * inv;
      Mout[(size_t)p * 1024 + r * 32 + 16 + l15] = t[ib][1][v] * inv;
    }
  if (lane == 0) Sout[p] = Sin[2 * p] + Sin[2 * p + 1] + __logf(mx);
}

// ---------------------------------------------------------------------------
// Kernel 4: alpha_0 = pi0 + ll_0 ; loss = -(log(sum(exp(alpha0-mz) @ M)) + mz + S)
// ---------------------------------------------------------------------------
__global__ __launch_bounds__(32) void final_kernel(
    const float* __restrict__ Y, const float* __restrict__ dT,
    const float* __restrict__ wsW, const float* __restrict__ wsPi0,
    const float* __restrict__ wsGa, const float* __restrict__ wsGb,
    const float* __restrict__ wsCst, const float* __restrict__ Mfin,
    const float* __restrict__ Sfin, float* __restrict__ outLoss) {
  const int lane = threadIdx.x & 31;
  float q = wsCst[lane];
  for (int d = 0; d < D_DIM; ++d) {
    float y = Y[d];
    q += y * y * wsW[d * 32 + lane] + y * wsW[(64 + d) * 32 + lane];
  }
  float dt0 = dT[0];
  q += logf(dt0) * (wsGa[lane] - 1.f) - dt0 * wsGb[lane];
  float az = wsPi0[lane] + q;
  float mz = redMax32(az);
  float v0 = __expf(az - mz);
  float vj = 0.f;
  for (int i = 0; i < 32; ++i) vj += __shfl(v0, i, 32) * Mfin[i * 32 + lane];
  float r = redSum32(vj);
  if (lane == 0) *outLoss = -(logf(r) + mz + Sfin[0]);
}

// ---------------------------------------------------------------------------
// Kernel 5: h = Y (identity embedding), float4 copy
// ---------------------------------------------------------------------------
__global__ void copy_kernel(const float4* __restrict__ src,
                            float4* __restrict__ dst, int n4) {
  int i = blockIdx.x * blockDim.x + threadIdx.x;
  if (i < n4) dst[i] = src[i];
}

// ---------------------------------------------------------------------------
extern "C" void kernel_launch(void* const* d_in, const int* in_sizes, int n_in,
                              void* d_out, int out_size, void* d_ws,
                              size_t ws_size, hipStream_t stream) {
  const float* Y = (const float*)d_in[0];
  const float* dT = (const float*)d_in[1];
  const float* A = (const float*)d_in[2];
  const float* log_ab = (const float*)d_in[3];
  const float* mu = (const float*)d_in[4];
  const float* log_sigma = (const float*)d_in[5];
  const float* pi = (const float*)d_in[6];
  float* out = (float*)d_out;

  float* ws = (float*)d_ws;
  float* wsW = ws;                 // 4096
  float* wsExpA = ws + 4096;       // 1024
  float* wsPi0 = ws + 5120;        // 32
  float* wsGa = ws + 5152;         // 32
  float* wsGb = ws + 5184;         // 32
  float* wsCst = ws + 5216;        // 32
  float* wsS0 = ws + 5248;         // P_CH
  float* wsS1 = wsS0 + P_CH;       // P_CH
  float* wsM0 = ws + 16384;        // P_CH * 1024
  float* wsM1 = wsM0 + (size_t)P_CH * 1024;  // (P_CH/2) * 1024

  precompute_kernel<<<1, 32, 0, stream>>>(A, log_ab, mu, log_sigma, pi, wsW,
                                          wsExpA, wsPi0, wsGa, wsGb, wsCst);

  chunk_kernel<<<P_CH / WPB, WPB * 32, 0, stream>>>(Y, dT, wsW, wsExpA, wsGa,
                                                    wsGb, wsCst, wsM0, wsS0);

  // pairwise tree reduce: 4096 -> 1
  const float* mi = wsM0;
  const float* si = wsS0;
  float* mo = wsM1;
  float* so = wsS1;
  int n = P_CH;
  while (n > 1) {
    int n2 = n >> 1;
    reduce_kernel<<<n2, 32, 0, stream>>>(mi, si, mo, so);
    const float* tm = mi; const float* ts = si;
    mi = mo; si = so;
    mo = (float*)tm; so = (float*)ts;
    n = n2;
  }

  final_kernel<<<1, 32, 0, stream>>>(Y, dT, wsW, wsPi0, wsGa, wsGb, wsCst, mi,
                                     si, out + (size_t)T_TOT * D_DIM);

  const int n4 = T_TOT * D_DIM / 4;
  copy_kernel<<<(n4 + 255) / 256, 256, 0, stream>>>((const float4*)Y,
                                                    (float4*)out, n4);
}